// RNNLayer_16381005267362
// MI455X (gfx1250) — compile-verified
//
#include <hip/hip_runtime.h>
#include <hip/hip_bf16.h>
#include <math.h>

typedef __attribute__((ext_vector_type(2))) float v2f;
typedef __attribute__((ext_vector_type(8))) float v8f;

constexpr int T = 256, B = 128, I = 1024, H = 1024;

// ---------------------------------------------------------------------------
// Phase 1: y[t*B+b][n] = sum_i x[t*B+b][i] * Wx[n][i] + bx[n]
// M = T*B = 32768, N = H = 1024, K = I = 1024.
// One wave computes a 32x64 output tile (2 M-tiles x 4 N-tiles) with
// V_WMMA_F32_16X16X4_F32: per K-step, 6 b64 loads feed 8 WMMAs.
// A (16x4 f32): lane<16 -> (M=lane, K=kk+0..1), lane>=16 -> (M=lane-16, K=kk+2..3)
//   => float2 load at x[row*I + kk + 2*(lane/16)]
// B (4x16 f32): lane<16 -> (K=kk+0..1, N=lane), lane>=16 -> (K=kk+2..3, N=lane-16)
//   => float2 load at Wx[col*I + kk + 2*(lane/16)]   (k contiguous => Wx^T free)
// ---------------------------------------------------------------------------
__global__ __launch_bounds__(256) void rnn_xproj_kernel(
    const float* __restrict__ x, const float* __restrict__ Wx,
    const float* __restrict__ bx, float* __restrict__ y)
{
    const int wave   = (int)((blockIdx.x * blockDim.x + threadIdx.x) >> 5);
    const int lane   = (int)(threadIdx.x & 31);
    const int half   = lane >> 4;       // 0 or 1
    const int lanelo = lane & 15;

    const int nStrips = H / 64;              // 16
    const int mBase   = (wave / nStrips) * 32;
    const int nBase   = (wave % nStrips) * 64;

    const float* aP0 = x  + (size_t)(mBase + lanelo) * I + 2 * half;
    const float* aP1 = aP0 + (size_t)16 * I;
    const float* bP  = Wx + (size_t)(nBase + lanelo) * I + 2 * half;

    v8f a0n0 = {}, a0n1 = {}, a0n2 = {}, a0n3 = {};
    v8f a1n0 = {}, a1n1 = {}, a1n2 = {}, a1n3 = {};

    #pragma unroll 2
    for (int kk = 0; kk < I; kk += 4) {
        v2f a0  = *(const v2f*)(aP0 + kk);
        v2f a1  = *(const v2f*)(aP1 + kk);
        v2f bb0 = *(const v2f*)(bP + kk);
        v2f bb1 = *(const v2f*)(bP + (size_t)16 * I + kk);
        v2f bb2 = *(const v2f*)(bP + (size_t)32 * I + kk);
        v2f bb3 = *(const v2f*)(bP + (size_t)48 * I + kk);
        a0n0 = __builtin_amdgcn_wmma_f32_16x16x4_f32(false, a0, false, bb0, (short)0, a0n0, false, false);
        a1n0 = __builtin_amdgcn_wmma_f32_16x16x4_f32(false, a1, false, bb0, (short)0, a1n0, false, false);
        a0n1 = __builtin_amdgcn_wmma_f32_16x16x4_f32(false, a0, false, bb1, (short)0, a0n1, false, false);
        a1n1 = __builtin_amdgcn_wmma_f32_16x16x4_f32(false, a1, false, bb1, (short)0, a1n1, false, false);
        a0n2 = __builtin_amdgcn_wmma_f32_16x16x4_f32(false, a0, false, bb2, (short)0, a0n2, false, false);
        a1n2 = __builtin_amdgcn_wmma_f32_16x16x4_f32(false, a1, false, bb2, (short)0, a1n2, false, false);
        a0n3 = __builtin_amdgcn_wmma_f32_16x16x4_f32(false, a0, false, bb3, (short)0, a0n3, false, false);
        a1n3 = __builtin_amdgcn_wmma_f32_16x16x4_f32(false, a1, false, bb3, (short)0, a1n3, false, false);
    }

    const float bias0 = bx[nBase + lanelo];
    const float bias1 = bx[nBase + lanelo + 16];
    const float bias2 = bx[nBase + lanelo + 32];
    const float bias3 = bx[nBase + lanelo + 48];

    #pragma unroll
    for (int r = 0; r < 8; ++r) {
        const int row0 = mBase + r + 8 * half;      // C/D layout: VGPR r, lane-half
        float* o0 = y + (size_t)row0 * H + nBase + lanelo;
        o0[0]  = a0n0[r] + bias0;
        o0[16] = a0n1[r] + bias1;
        o0[32] = a0n2[r] + bias2;
        o0[48] = a0n3[r] + bias3;
        float* o1 = o0 + (size_t)16 * H;
        o1[0]  = a1n0[r] + bias0;
        o1[16] = a1n1[r] + bias1;
        o1[32] = a1n2[r] + bias2;
        o1[48] = a1n3[r] + bias3;
    }
}

// ---------------------------------------------------------------------------
// Phase 2 (one launch per timestep, graph-captured):
//   yt[b][n] = tanh( yt[b][n] + sum_k hprev[b][k] * Wh[n][k] )
// yt holds xw_t on entry (written by phase 1), holds h_t on exit.
// M = 128, N = K = 1024 -> 512 16x16 tiles. 4-way K-split: 4 waves per tile
// (K chunk = 256 -> two 32-step WMMA chains), partial sums reduced in LDS,
// K-slice-0 wave applies xw + tanh and stores. 2 tiles per 256-thread block.
// ---------------------------------------------------------------------------
__global__ __launch_bounds__(256) void rnn_step_kernel(
    const float* __restrict__ hprev, const float* __restrict__ Wh,
    float* __restrict__ yt)
{
    __shared__ float red[2][3][32][8];          // [tileInBlk][kslice-1][lane][r]

    const int waveInBlk = (int)(threadIdx.x >> 5);  // 0..7
    const int lane      = (int)(threadIdx.x & 31);
    const int half      = lane >> 4;
    const int lanelo    = lane & 15;

    const int tib  = waveInBlk >> 2;            // tile within block: 0/1
    const int ks   = waveInBlk & 3;             // K slice: 0..3
    const int tile = (int)blockIdx.x * 2 + tib; // 0..511
    const int mBase = (tile & 7) * 16;          // B/16 = 8 M-tiles
    const int nBase = (tile >> 3) * 16;         // H/16 = 64 N-tiles
    const int k0   = ks * (H / 4);              // 256-wide K chunk

    const float* aP = hprev + (size_t)(mBase + lanelo) * H + k0 + 2 * half;
    const float* bP = Wh    + (size_t)(nBase + lanelo) * H + k0 + 2 * half;

    v8f acc0 = {}, acc1 = {};
    #pragma unroll 4
    for (int kk = 0; kk < H / 4; kk += 8) {
        v2f a0 = *(const v2f*)(aP + kk);
        v2f b0 = *(const v2f*)(bP + kk);
        v2f a1 = *(const v2f*)(aP + kk + 4);
        v2f b1 = *(const v2f*)(bP + kk + 4);
        acc0 = __builtin_amdgcn_wmma_f32_16x16x4_f32(false, a0, false, b0, (short)0, acc0, false, false);
        acc1 = __builtin_amdgcn_wmma_f32_16x16x4_f32(false, a1, false, b1, (short)0, acc1, false, false);
    }
    v8f acc = acc0 + acc1;

    if (ks != 0) {
        float* p = &red[tib][ks - 1][lane][0];
        #pragma unroll
        for (int r = 0; r < 8; ++r) p[r] = acc[r];
    }
    __syncthreads();
    if (ks == 0) {
        #pragma unroll
        for (int s = 0; s < 3; ++s) {
            const float* p = &red[tib][s][lane][0];
            #pragma unroll
            for (int r = 0; r < 8; ++r) acc[r] += p[r];
        }
        #pragma unroll
        for (int r = 0; r < 8; ++r) {
            const int row = mBase + r + 8 * half;
            float* o = yt + (size_t)row * H + nBase + lanelo;
            o[0] = tanhf(acc[r] + o[0]);
        }
    }
}

__global__ __launch_bounds__(256) void rnn_copy_kernel(
    const float* __restrict__ src, float* __restrict__ dst, int n)
{
    const int i = (int)(blockIdx.x * blockDim.x + threadIdx.x);
    if (i < n) dst[i] = src[i];
}

extern "C" void kernel_launch(void* const* d_in, const int* in_sizes, int n_in,
                              void* d_out, int out_size, void* d_ws, size_t ws_size,
                              hipStream_t stream) {
    (void)in_sizes; (void)n_in; (void)out_size; (void)d_ws; (void)ws_size;
    const float* x     = (const float*)d_in[0]; // [T, B, I]
    const float* state = (const float*)d_in[1]; // [B, H]
    const float* Wx    = (const float*)d_in[2]; // [H, I]
    const float* bx    = (const float*)d_in[3]; // [H]
    const float* Wh    = (const float*)d_in[4]; // [H, H]

    float* y      = (float*)d_out;              // [T, B, H]
    float* hfinal = y + (size_t)T * B * H;      // [B, H]

    // Phase 1: fill y with xW + bx (one wave per 32x64 tile).
    const int waves1  = (T * B / 32) * (H / 64);   // 16384 waves
    const int blocks1 = waves1 / 8;                // 8 waves per 256-thread block
    rnn_xproj_kernel<<<blocks1, 256, 0, stream>>>(x, Wx, bx, y);

    // Phase 2: sequential scan; y[t] is xw_t on input, h_t on output.
    // 512 tiles x 4 K-slices = 2048 waves = 256 blocks per step.
    for (int t = 0; t < T; ++t) {
        const float* hprev = (t == 0) ? state : (y + (size_t)(t - 1) * B * H);
        rnn_step_kernel<<<256, 256, 0, stream>>>(hprev, Wh, y + (size_t)t * B * H);
    }

    // h_final = y[T-1]
    rnn_copy_kernel<<<(B * H + 255) / 256, 256, 0, stream>>>(
        y + (size_t)(T - 1) * B * H, hfinal, B * H);
}